// DualLevelMemoryModule_69063074120416
// MI455X (gfx1250) — compile-verified
//
#include <hip/hip_runtime.h>
#include <hip/hip_bf16.h>

// Problem constants (from reference)
#define NB   16          // batch
#define NC   128         // feature channels
#define HH   256
#define WW   256
#define HMN  (HH * WW)   // 65536
#define KPK  8           // top-k peaks
#define NMSR 5           // NMS radius

// d_out flat layout (tuple return order)
#define OFF_FEAT 1048576            // reweighted: [0, 16*65536)
#define OFF_POS  1064960            // peak_feats: 16*8*128
#define OFF_PROB 1065216            // positions:  16*8*2
#define OFF_VAL  1065344            // probs: 128, valid: 128

// d_ws float layout: pr[128] @0, pc[128] @128, w=prob*valid[128] @256
#define WS_PR  0
#define WS_PC  128
#define WS_WV  256

typedef __attribute__((ext_vector_type(2))) float v2f;
typedef __attribute__((ext_vector_type(8))) float v8f;

// ---------------------------------------------------------------------------
// Kernel 1: NMS + global max + top-8 + softmax. One block per batch image.
// ---------------------------------------------------------------------------
__global__ __launch_bounds__(256) void peaks_kernel(const float* __restrict__ hm,
                                                    float* __restrict__ out,
                                                    float* __restrict__ ws) {
  const int b   = blockIdx.x;
  const int tid = threadIdx.x;
  const float* hb = hm + b * HMN;

  __shared__ float red[256];
  __shared__ float vmax[WW];
  __shared__ float tS[256 * KPK];
  __shared__ int   tI[256 * KPK];

  // ---- global max over the image ----
  float lm = -3.4e38f;
  for (int i = tid; i < HMN; i += 256) lm = fmaxf(lm, hb[i]);
  red[tid] = lm;
  __syncthreads();
  for (int s = 128; s > 0; s >>= 1) {
    if (tid < s) red[tid] = fmaxf(red[tid], red[tid + s]);
    __syncthreads();
  }
  const float gthr = red[0] * 0.05f;   // SCORE_THRESHOLD * gmax
  __syncthreads();

  // ---- separable 11x11 window max + per-thread top-8 (thread owns column) ----
  float ls[KPK]; int li[KPK];
#pragma unroll
  for (int j = 0; j < KPK; ++j) { ls[j] = -3.4e38f; li[j] = 0x7fffffff; }

  const int c = tid;
  const int clo = (c - NMSR < 0) ? 0 : c - NMSR;
  const int chi = (c + NMSR > WW - 1) ? WW - 1 : c + NMSR;
  for (int r = 0; r < HH; ++r) {
    // prefetch the row that enters the window next iteration
    if (r + NMSR + 1 < HH) __builtin_prefetch(hb + (r + NMSR + 1) * WW + c, 0, 3);
    // vertical pass: max over rows [r-5, r+5] for this column
    const int rlo = (r - NMSR < 0) ? 0 : r - NMSR;
    const int rhi = (r + NMSR > HH - 1) ? HH - 1 : r + NMSR;
    float vm = -3.4e38f;
    for (int rr = rlo; rr <= rhi; ++rr) vm = fmaxf(vm, hb[rr * WW + c]);
    vmax[c] = vm;
    __syncthreads();
    // horizontal pass from LDS
    float wm = -3.4e38f;
    for (int cc = clo; cc <= chi; ++cc) wm = fmaxf(wm, vmax[cc]);
    const float v  = hb[r * WW + c];
    const float sc = ((v >= wm) && (v >= gthr)) ? v : -1.0e9f;
    // insertion sort into descending top-8; strict '>' keeps lower idx on ties
    if (sc > ls[KPK - 1]) {
      int p = KPK - 1;
      while (p > 0 && sc > ls[p - 1]) { ls[p] = ls[p - 1]; li[p] = li[p - 1]; --p; }
      ls[p] = sc; li[p] = r * WW + c;
    }
    __syncthreads();
  }

#pragma unroll
  for (int j = 0; j < KPK; ++j) { tS[tid * KPK + j] = ls[j]; tI[tid * KPK + j] = li[j]; }
  __syncthreads();

  // ---- thread 0: merge 256 sorted top-8 lists (tie-break: lowest flat idx) ----
  if (tid == 0) {
    float fs[KPK]; int fi[KPK];
    for (int s = 0; s < KPK; ++s) {
      float bs = -3.4e38f; int bi = 0x7fffffff; int bj = 0;
      for (int j = 0; j < 256 * KPK; ++j) {
        const float scj = tS[j]; const int idj = tI[j];
        if (scj > bs || (scj == bs && idj < bi)) { bs = scj; bi = idj; bj = j; }
      }
      fs[s] = bs; fi[s] = bi;
      tS[bj] = -3.4e38f; tI[bj] = 0x7fffffff;
    }
    // softmax over the 8 scores (fs[0] is the max)
    const float m = fs[0];
    float e[KPK], sum = 0.f;
    for (int j = 0; j < KPK; ++j) { e[j] = __expf(fs[j] - m); sum += e[j]; }
    const float inv = 1.0f / sum;
    for (int j = 0; j < KPK; ++j) {
      const float rowf = (float)(fi[j] / WW);
      const float colf = (float)(fi[j] % WW);
      const float p  = e[j] * inv;
      const float vf = (fs[j] > -1.0e8f) ? 1.0f : 0.0f;
      out[OFF_POS  + (b * KPK + j) * 2 + 0] = rowf;
      out[OFF_POS  + (b * KPK + j) * 2 + 1] = colf;
      out[OFF_PROB + b * KPK + j] = p;
      out[OFF_VAL  + b * KPK + j] = vf;
      ws[WS_PR + b * KPK + j] = rowf;
      ws[WS_PC + b * KPK + j] = colf;
      ws[WS_WV + b * KPK + j] = p * vf;
    }
  }
}

// ---------------------------------------------------------------------------
// Kernel 2: bilinear feature sampling at peaks. block = (b,k), thread = channel
// ---------------------------------------------------------------------------
__global__ __launch_bounds__(NC) void sample_kernel(const float* __restrict__ fm,
                                                    const float* __restrict__ ws,
                                                    float* __restrict__ out) {
  const int b  = blockIdx.x / KPK;
  const int k  = blockIdx.x % KPK;
  const int ch = threadIdx.x;
  const float rr = ws[WS_PR + b * KPK + k];
  const float cc = ws[WS_PC + b * KPK + k];
  const float r0 = floorf(rr), c0 = floorf(cc);
  const float wr = rr - r0,    wc = cc - c0;
  int r0i = (int)r0, c0i = (int)c0;
  int r1i = r0i + 1, c1i = c0i + 1;
  r0i = min(max(r0i, 0), HH - 1); r1i = min(max(r1i, 0), HH - 1);
  c0i = min(max(c0i, 0), WW - 1); c1i = min(max(c1i, 0), WW - 1);
  const float* fb = fm + ((size_t)b * NC + ch) * (size_t)HMN;
  const float f00 = fb[r0i * WW + c0i];
  const float f01 = fb[r0i * WW + c1i];
  const float f10 = fb[r1i * WW + c0i];
  const float f11 = fb[r1i * WW + c1i];
  out[OFF_FEAT + (b * KPK + k) * NC + ch] =
      f00 * (1.f - wr) * (1.f - wc) + f01 * (1.f - wr) * wc +
      f10 * wr * (1.f - wc) + f11 * wr * wc;
}

// ---------------------------------------------------------------------------
// Kernel 3 (fused): rank-8 Gaussian weight map as a WMMA GEMM + in-block
// max-normalization + heatmap reweight. One 1024-thread block (32 waves) per
// batch; wave w owns M-tile (w/2) and N-tiles [(w&1)*8, (w&1)*8+8).
//
//   wm[m,n] = sum_k (w_k*exp(-(m-pr_k)^2/50)) * exp(-(n-pc_k)^2/50)
//
// A (16x4 f32) frag: lanes 0-15 -> M=lane, dwords hold K={0,1};
//                    lanes 16-31 -> M=lane-16, dwords hold K={2,3}. B mirrors.
// The A fragment is built once per wave and reused by all 16 WMMAs.
// C/D layout: dword i -> M = mr0 + i + 8*(lane>=16), N = nc0 + (lane&15).
// ---------------------------------------------------------------------------
__device__ inline v8f gauss_tile(const v2f afrag[2], const float pck[4],
                                 int n, float inv2s2) {
  v8f acc = {};
#pragma unroll
  for (int h = 0; h < 2; ++h) {
    const float dn0 = (float)n - pck[h * 2 + 0];
    const float dn1 = (float)n - pck[h * 2 + 1];
    v2f bv;
    bv.x = __expf(-dn0 * dn0 * inv2s2);
    bv.y = __expf(-dn1 * dn1 * inv2s2);
    acc = __builtin_amdgcn_wmma_f32_16x16x4_f32(
        /*neg_a=*/false, afrag[h], /*neg_b=*/false, bv,
        /*c_mod=*/(short)0, acc, /*reuse_a=*/false, /*reuse_b=*/false);
  }
  return acc;
}

__global__ __launch_bounds__(1024) void reweight_kernel(const float* __restrict__ hm,
                                                        const float* __restrict__ ws,
                                                        float* __restrict__ out) {
  const int b    = blockIdx.x;
  const int tid  = threadIdx.x;
  const int wave = tid >> 5;          // 0..31
  const int lane = tid & 31;
  const int mr0  = (wave >> 1) * 16;  // M-tile origin
  const int ntb  = (wave & 1) * 8;    // first N-tile index (0 or 8)
  const int lm15 = lane & 15;
  const int koff = (lane >> 4) * 2;   // this half-wave holds K = koff, koff+1 per dword
  const float inv2s2 = 1.0f / (2.0f * 5.0f * 5.0f);  // 1/(2*SIGMA^2)

  // Per-lane peak parameters for K slots {koff, koff+1} of each K-half.
  float prk[4], pck[4], wvk[4];
#pragma unroll
  for (int h = 0; h < 2; ++h) {
#pragma unroll
    for (int d = 0; d < 2; ++d) {
      const int k = h * 4 + koff + d;
      prk[h * 2 + d] = ws[WS_PR + b * KPK + k];
      pck[h * 2 + d] = ws[WS_PC + b * KPK + k];
      wvk[h * 2 + d] = ws[WS_WV + b * KPK + k];
    }
  }

  // A fragment: reused by all 16 WMMAs of this wave.
  const int m = mr0 + lm15;
  v2f afrag[2];
#pragma unroll
  for (int h = 0; h < 2; ++h) {
    const float dm0 = (float)m - prk[h * 2 + 0];
    const float dm1 = (float)m - prk[h * 2 + 1];
    afrag[h].x = wvk[h * 2 + 0] * __expf(-dm0 * dm0 * inv2s2);
    afrag[h].y = wvk[h * 2 + 1] * __expf(-dm1 * dm1 * inv2s2);
  }

  // ---- phase 1: per-wave tile sweep for the weight-map max (values >= 0) ----
  float lmax = 0.0f;
#pragma unroll
  for (int j = 0; j < 8; ++j) {
    const int n = (ntb + j) * 16 + lm15;
    const v8f acc = gauss_tile(afrag, pck, n, inv2s2);
#pragma unroll
    for (int i = 0; i < 8; ++i) lmax = fmaxf(lmax, acc[i]);
  }

  __shared__ float smax[1024];
  smax[tid] = lmax;
  __syncthreads();
  for (int s = 512; s > 0; s >>= 1) {
    if (tid < s) smax[tid] = fmaxf(smax[tid], smax[tid + s]);
    __syncthreads();
  }
  const float invn = 1.0f / (smax[0] + 1e-8f);

  // ---- phase 2: recompute tiles (cheap; A frag reused) and write output ----
  const int mb = mr0 + ((lane >> 4) * 8);
#pragma unroll
  for (int j = 0; j < 8; ++j) {
    const int n = (ntb + j) * 16 + lm15;
    const v8f acc = gauss_tile(afrag, pck, n, inv2s2);
#pragma unroll
    for (int i = 0; i < 8; ++i) {
      const int idx = b * HMN + (mb + i) * WW + n;
      out[idx] = hm[idx] * (0.5f + 0.5f * acc[i] * invn);
    }
  }
}

// ---------------------------------------------------------------------------
extern "C" void kernel_launch(void* const* d_in, const int* in_sizes, int n_in,
                              void* d_out, int out_size, void* d_ws, size_t ws_size,
                              hipStream_t stream) {
  (void)in_sizes; (void)n_in; (void)out_size; (void)ws_size;
  const float* hm = (const float*)d_in[0];   // heatmap (16,1,256,256) f32
  const float* fm = (const float*)d_in[1];   // feature_map (16,128,256,256) f32
  float* out = (float*)d_out;
  float* ws  = (float*)d_ws;

  peaks_kernel   <<<NB,       256,  0, stream>>>(hm, out, ws);
  sample_kernel  <<<NB * KPK, NC,   0, stream>>>(fm, ws, out);
  reweight_kernel<<<NB,       1024, 0, stream>>>(hm, ws, out);
}